// SlotElmanCell_30193620091474
// MI455X (gfx1250) — compile-verified
//
#include <hip/hip_runtime.h>

// SlotElmanCell scan for MI455X (gfx1250, wave32).
//
// Roofline: ~563 MB total traffic, 537 MB of which is the h-state store
// stream -> pure store-bandwidth-bound (~24 us @ 23.3 TB/s). Compute is
// ~0.9 GFLOP, >100x under the memory floor, and the recurrence weights are
// element-wise per (d,s), so there is no shared contraction for WMMA.
//
// Mapping: one wave per (b,d) chain (2048 waves), 2 state slots per lane,
// 256 B/wave/step coalesced non-temporal b64 stores for h.
// CDNA5 async paths (ASYNCcnt):
//  - x/z staged per-block into LDS via double-buffered
//    GLOBAL_LOAD_ASYNC_TO_LDS_B128 (hot-loop loads become LDS broadcasts).
//  - out accumulated per-block in LDS and drained in 2 KB tiles via
//    GLOBAL_STORE_ASYNC_FROM_LDS_B128 (262K requests instead of 2M 4 B
//    stores, which would otherwise double the store-request rate vs h).

typedef __attribute__((ext_vector_type(2))) float f32x2;

namespace {
constexpr int T  = 1024;
constexpr int BB = 2;
constexpr int D  = 1024;
constexpr int S  = 64;
constexpr int CHAINS = BB * D;            // 2048 independent recurrences
constexpr int WPB = 8;                    // waves (= chains) per block
constexpr int THREADS = WPB * 32;         // 256 threads
constexpr int TBD = T * BB * D;           // elements in `output`
constexpr int PLANE = CHAINS * S;         // elements in one h[t] plane
constexpr int CT = 64;                    // timesteps per staged chunk
constexpr int NCHUNK = T / CT;            // 16 chunks
}

__global__ __launch_bounds__(THREADS)
void slot_elman_scan(const float* __restrict__ x,            // [T,BB,D]
                     const float* __restrict__ z,            // [T,BB,D]
                     const float* __restrict__ h0,           // [BB,D,S]
                     const float* __restrict__ decay_logits, // [D,S]
                     const float* __restrict__ Bmat,         // [D,S]
                     const float* __restrict__ Cvec,         // [S]
                     float* __restrict__ out,                // [T,BB,D]
                     float* __restrict__ hout)               // [T+1,BB,D,S]
{
  // Double-buffered chunk tiles: [buf][t_local * WPB + d_local]
  __shared__ __align__(16) float sx[2][CT * WPB];  // x stage-in
  __shared__ __align__(16) float sz[2][CT * WPB];  // z stage-in
  __shared__ __align__(16) float so[2][CT * WPB];  // out stage-out

  const int widx  = threadIdx.x >> 5;        // wave index in block = d_local
  const int lane  = threadIdx.x & 31;
  const int chain = blockIdx.x * WPB + widx; // (b,d) chain id
  const int b  = chain >> 10;                // chain / D   (per-block constant)
  const int d  = chain & (D - 1);            // chain % D
  const int d_base = (blockIdx.x * WPB) & (D - 1);
  const int s0 = lane << 1;                  // this lane owns slots s0, s0+1

  // Async global->LDS: stage one CT x 32 B x/z tile (4 b128 instrs per stream).
  auto stage_in = [&](int buf, int t0) {
    const int tr0  = lane >> 1;              // 2 lanes per 32 B row
    const int half = (lane & 1) * 4;         // which 16 B of the row
#pragma unroll
    for (int q = 0; q < CT / 16; ++q) {
      const int trow = tr0 + q * 16;
      const float* gx = x + ((size_t)(t0 + trow) * BB + b) * D + d_base + half;
      const float* gz = z + ((size_t)(t0 + trow) * BB + b) * D + d_base + half;
      const unsigned lx = (unsigned)(uintptr_t)&sx[buf][trow * WPB + half];
      const unsigned lz = (unsigned)(uintptr_t)&sz[buf][trow * WPB + half];
      asm volatile("global_load_async_to_lds_b128 %0, %1, off"
                   :: "v"(lx), "v"(gx) : "memory");
      asm volatile("global_load_async_to_lds_b128 %0, %1, off"
                   :: "v"(lz), "v"(gz) : "memory");
    }
  };

  // Async LDS->global: drain one CT x 32 B out tile (4 b128 instrs).
  auto drain_out = [&](int buf, int t0) {
    const int tr0  = lane >> 1;
    const int half = (lane & 1) * 4;
#pragma unroll
    for (int q = 0; q < CT / 16; ++q) {
      const int trow = tr0 + q * 16;
      float* go = out + ((size_t)(t0 + trow) * BB + b) * D + d_base + half;
      const unsigned lo = (unsigned)(uintptr_t)&so[buf][trow * WPB + half];
      asm volatile("global_store_async_from_lds_b128 %0, %1, off"
                   :: "v"(go), "v"(lo) : "memory");
    }
  };

  // Per-chain constants (one-time loads, negligible traffic).
  const f32x2 dl = *(const f32x2*)(decay_logits + d * S + s0);
  f32x2 a;
  a.x = 1.0f / (1.0f + expf(-dl.x));         // sigmoid(decay_logits)
  a.y = 1.0f / (1.0f + expf(-dl.y));
  const f32x2 Bl = *(const f32x2*)(Bmat + d * S + s0);
  const f32x2 Cl = *(const f32x2*)(Cvec + s0);

  // Initial state; emit h[0] plane (harness poisons d_out -> write all of it).
  f32x2 h = *(const f32x2*)(h0 + (size_t)chain * S + s0);
  __builtin_nontemporal_store(h, (f32x2*)(hout + (size_t)chain * S + s0));

  float* hp = hout + (size_t)PLANE + (size_t)chain * S + s0;   // h[1] plane

  // Prime the pipeline with chunk 0.
  if (widx == 0) stage_in(0, 0);
  asm volatile("s_wait_asynccnt 0x0" ::: "memory");  // no-op for non-stagers
  __syncthreads();

  for (int c = 0; c < NCHUNK; ++c) {
    const int cur = c & 1;
    if (widx == 0) {
      // Overlap with compute: prefetch chunk c+1, drain out-tile of chunk c-1
      // (its LDS writes were made visible by the previous barrier).
      if (c + 1 < NCHUNK) stage_in(cur ^ 1, (c + 1) * CT);
      if (c > 0)          drain_out(cur ^ 1, (c - 1) * CT);
    }

#pragma unroll 4
    for (int tl = 0; tl < CT; ++tl) {
      const int t = c * CT + tl;
      // volatile: these LDS cells are written only by the async engine.
      const float xt = *(volatile const float*)&sx[cur][tl * WPB + widx];
      const float zt = *(volatile const float*)&sz[cur][tl * WPB + widx];

      // h = sigmoid(decay)*h + B*x_t   (element-wise recurrence)
      h.x = fmaf(a.x, h.x, Bl.x * xt);
      h.y = fmaf(a.y, h.y, Bl.y * xt);

      // out = (h . C) * silu(z_t): per-lane partial + wave32 butterfly
      float p = fmaf(h.x, Cl.x, h.y * Cl.y);
#pragma unroll
      for (int off = 16; off > 0; off >>= 1)
        p += __shfl_xor(p, off, 32);

      if (lane == 0) {
        const float sig = 1.0f / (1.0f + expf(-zt));
        // volatile: read only by the async drain engine.
        *(volatile float*)&so[cur][tl * WPB + widx] = p * zt * sig;
      }

      // Dominant traffic: 256 B/wave/step, contiguous, never re-read ->
      // NT hint keeps the 537 MB stream from churning the 192 MB L2.
      __builtin_nontemporal_store(h, (f32x2*)(hp + (size_t)t * PLANE));
    }

    // Guarantees before buffer swap: next x/z tile landed; out tile of the
    // previous chunk fully drained (safe to overwrite next iteration).
    asm volatile("s_wait_asynccnt 0x0" ::: "memory");
    __syncthreads();
  }

  // Drain the final out tile (made visible by the last barrier).
  if (widx == 0) drain_out((NCHUNK - 1) & 1, (NCHUNK - 1) * CT);
  asm volatile("s_wait_asynccnt 0x0" ::: "memory");
}

extern "C" void kernel_launch(void* const* d_in, const int* in_sizes, int n_in,
                              void* d_out, int out_size, void* d_ws, size_t ws_size,
                              hipStream_t stream) {
  const float* x  = (const float*)d_in[0];
  const float* zz = (const float*)d_in[1];
  const float* h0 = (const float*)d_in[2];
  const float* dl = (const float*)d_in[3];
  const float* Bm = (const float*)d_in[4];
  const float* Cv = (const float*)d_in[5];
  float* out  = (float*)d_out;          // output: [T,BB,D]
  float* hout = out + TBD;              // h:      [T+1,BB,D,S]

  dim3 grid(CHAINS / WPB);              // 256 blocks
  dim3 block(THREADS);                  // 256 threads = 8 wave32
  slot_elman_scan<<<grid, block, 0, stream>>>(x, zz, h0, dl, Bm, Cv, out, hout);
}